// XToQ_39754217292542
// MI455X (gfx1250) — compile-verified
//
#include <hip/hip_runtime.h>
#include <math.h>

// ---------------------------------------------------------------------------
// MI455X (gfx1250) implementation.
//
// Roofline: after folding K_w/V_w onto the small side of the attention
// (Kh/Vh never materialized, LSTM collapsed to its single surviving step),
// the work is ~1.1 GFLOP against one 64 MB streaming operand (hidden) ->
// memory-bound at 23.3 TB/s (~3 us).  fp32 is therefore free vs bf16, so all
// GEMMs use the CDNA5 fp32 matrix unit V_WMMA_F32_16X16X4_F32 at full
// reference fidelity.  attn (4 MB) + weights stay resident in the 192 MB L2.
// Tile staging uses GLOBAL_LOAD_ASYNC_TO_LDS_B128 (ASYNCcnt) when the
// toolchain exposes it, plus global_prefetch for the next tile.
// ---------------------------------------------------------------------------

#define S_DIM  2048
#define B_DIM  32
#define H_DIM  256
#define NX_DIM 16
#define BH     (B_DIM * H_DIM)   // 8192 floats between consecutive s rows

typedef __attribute__((ext_vector_type(2))) float f32x2;
typedef __attribute__((ext_vector_type(8))) float f32x8;
typedef int v4i __attribute__((__vector_size__(4 * sizeof(int))));

#if defined(__AMDGCN__) &&                                                    \
    __has_builtin(__builtin_amdgcn_global_load_async_to_lds_b128) &&          \
    __has_builtin(__builtin_amdgcn_s_wait_asynccnt)
#define HAVE_ASYNC_LDS 1
#else
#define HAVE_ASYNC_LDS 0
#endif

#if HAVE_ASYNC_LDS
typedef __attribute__((address_space(1))) v4i* gptr_v4i;
typedef __attribute__((address_space(3))) v4i* lptr_v4i;
#endif

// memory -> LDS tile copy, one 16B chunk per lane
__device__ __forceinline__ void cp_b128(float* lds_dst, const float* gsrc) {
#if HAVE_ASYNC_LDS
  __builtin_amdgcn_global_load_async_to_lds_b128((gptr_v4i)gsrc,
                                                 (lptr_v4i)lds_dst, 0, 0);
#else
  *(float4*)lds_dst = *(const float4*)gsrc;
#endif
}

// drain this wave's async copies, then workgroup barrier
__device__ __forceinline__ void cp_wait_barrier() {
#if HAVE_ASYNC_LDS
  __builtin_amdgcn_s_wait_asynccnt(0);
#endif
  __syncthreads();
}

// D(16x16,f32) = A(16x4,f32) * B(4x16,f32) + C
// A frag: lanes 0-15 -> M=lane, k={k0,k0+1}; lanes 16-31 -> M=lane-16, k={k0+2,k0+3}
// B frag: lanes 0-15 -> N=lane, k={k0,k0+1}; lanes 16-31 -> N=lane-16, k={k0+2,k0+3}
// D:      vgpr r, lane L -> M = r + 8*(L>>4), N = L&15
__device__ __forceinline__ f32x8 wmma4(f32x2 a, f32x2 b, f32x8 c) {
  return __builtin_amdgcn_wmma_f32_16x16x4_f32(false, a, false, b, (short)0, c,
                                               false, false);
}

// ---------------------------------------------------------------------------
// xK[b,n,h'] = sum_h x[b,n,h] * K_w[h,h']      (M=16 queries, N=256, K=256)
// grid (4, B), 128 threads (4 waves x one 16-wide N tile)
// ---------------------------------------------------------------------------
__global__ void k_xk(const float* __restrict__ x, const float* __restrict__ Kw,
                     float* __restrict__ xK) {
  const int b     = blockIdx.y;
  const int h0    = blockIdx.x * 64;
  const int lane  = threadIdx.x & 31;
  const int wv    = threadIdx.x >> 5;
  const int lmod  = lane & 15;
  const int lhalf = lane >> 4;
  const int col   = h0 + wv * 16 + lmod;

  f32x8 acc = {};
  for (int k0 = 0; k0 < H_DIM; k0 += 4) {
    const int bk = k0 + 2 * lhalf;
    f32x2 a = *(const f32x2*)(x + (size_t)(b * NX_DIM + lmod) * H_DIM + bk);
    f32x2 bb;
    bb.x = Kw[(size_t)bk * H_DIM + col];
    bb.y = Kw[(size_t)(bk + 1) * H_DIM + col];
    acc = wmma4(a, bb, acc);
  }
#pragma unroll
  for (int r = 0; r < 8; ++r) {
    const int m = r + 8 * lhalf;
    xK[(size_t)(b * NX_DIM + m) * H_DIM + col] = acc[r];
  }
}

// ---------------------------------------------------------------------------
// xb[b,n] = x[b,n,:] . K_b        grid (B), 128 threads
// ---------------------------------------------------------------------------
__global__ void k_xb(const float* __restrict__ x, const float* __restrict__ Kb,
                     float* __restrict__ xb) {
  __shared__ float red[128];
  const int b   = blockIdx.x;
  const int t   = threadIdx.x;
  const int n   = t >> 3;
  const int seg = t & 7;
  float s = 0.f;
  for (int i = 0; i < 32; ++i) {
    const int h = seg * 32 + i;
    s += x[(size_t)(b * NX_DIM + n) * H_DIM + h] * Kb[h];
  }
  red[t] = s;
  __syncthreads();
  if (seg == 0) {
    float tot = 0.f;
    for (int i = 0; i < 8; ++i) tot += red[n * 8 + i];
    xb[b * NX_DIM + n] = tot;
  }
}

// ---------------------------------------------------------------------------
// scores[b,n,s] = xK[b,n,:] . hidden[s,b,:] + xb[b,n]
// grid (S/64, B), 128 threads.  hidden staged in LDS in two 64x128 k-slices
// via async-to-LDS.  LDS row stride 132 (%64 == 4) -> the 64 dwords touched
// by one paired-f32x2 B-fragment read hit 64 distinct banks.
// ---------------------------------------------------------------------------
__global__ void k_scores(const float* __restrict__ hidden,
                         const float* __restrict__ xK,
                         const float* __restrict__ xb,
                         float* __restrict__ scores) {
  __shared__ float lds_h[64 * 132];
  const int b     = blockIdx.y;
  const int s0    = blockIdx.x * 64;
  const int t     = threadIdx.x;
  const int lane  = t & 31;
  const int wv    = t >> 5;
  const int lmod  = lane & 15;
  const int lhalf = lane >> 4;

  // pull the second k-slice (consumed in iteration kh=1) toward the caches
  {
    const float* nxt =
        hidden + (size_t)(s0 + (t >> 1)) * BH + b * H_DIM + 128 + (t & 1) * 32;
    __builtin_prefetch(nxt, 0, 1);
  }

  f32x8 acc = {};
  for (int kh = 0; kh < 2; ++kh) {
    __syncthreads();
    // stage 64 s-rows x 128 k floats, coalesced async b128 copies
#pragma unroll
    for (int i = 0; i < 16; ++i) {
      const int idx = (i * 128 + t) * 4;
      const int si  = idx >> 7;
      const int kk  = idx & 127;
      cp_b128(&lds_h[si * 132 + kk],
              hidden + (size_t)(s0 + si) * BH + b * H_DIM + kh * 128 + kk);
    }
    cp_wait_barrier();
#pragma unroll 4
    for (int k0 = 0; k0 < 128; k0 += 4) {
      const int bk = k0 + 2 * lhalf;
      f32x2 a = *(const f32x2*)(xK + (size_t)(b * NX_DIM + lmod) * H_DIM +
                                kh * 128 + bk);
      f32x2 bb = *(const f32x2*)(&lds_h[(wv * 16 + lmod) * 132 + bk]);
      acc = wmma4(a, bb, acc);
    }
  }
#pragma unroll
  for (int r = 0; r < 8; ++r) {
    const int m = r + 8 * lhalf;
    scores[(size_t)(b * NX_DIM + m) * S_DIM + s0 + wv * 16 + lmod] =
        acc[r] + xb[b * NX_DIM + m];
  }
}

// ---------------------------------------------------------------------------
// softmax over S for each of B*NX rows.  grid (B*NX), 256 threads.
// ---------------------------------------------------------------------------
__global__ void k_softmax(float* __restrict__ attn) {
  __shared__ float red[256];
  const int row = blockIdx.x;
  const int t   = threadIdx.x;
  float* p = attn + (size_t)row * S_DIM;

  float v[8];
  float m = -3.4e38f;
#pragma unroll
  for (int i = 0; i < 8; ++i) {
    v[i] = p[t + i * 256];
    m = fmaxf(m, v[i]);
  }
  red[t] = m;
  __syncthreads();
  for (int off = 128; off > 0; off >>= 1) {
    if (t < off) red[t] = fmaxf(red[t], red[t + off]);
    __syncthreads();
  }
  m = red[0];
  __syncthreads();

  float s = 0.f;
#pragma unroll
  for (int i = 0; i < 8; ++i) {
    v[i] = __expf(v[i] - m);
    s += v[i];
  }
  red[t] = s;
  __syncthreads();
  for (int off = 128; off > 0; off >>= 1) {
    if (t < off) red[t] += red[t + off];
    __syncthreads();
  }
  const float inv = 1.f / red[0];
#pragma unroll
  for (int i = 0; i < 8; ++i) p[t + i * 256] = v[i] * inv;
}

// ---------------------------------------------------------------------------
// aH[b,n,h'] = sum_s attn[b,n,s] * hidden[s,b,h']   (K = 2048)
// grid (4, B), 128 threads.  attn tile 16x128 (stride 132), hidden tile
// 128x64 (stride 72, %64==8 -> each scalar B-fragment dword read hits 32
// distinct banks).  Tiles staged via async-to-LDS; next s-tile prefetched.
// ---------------------------------------------------------------------------
__global__ void k_aH(const float* __restrict__ hidden,
                     const float* __restrict__ attn,
                     float* __restrict__ aH) {
  __shared__ float lds_a[16 * 132];
  __shared__ float lds_h[128 * 72];
  const int b     = blockIdx.y;
  const int h0    = blockIdx.x * 64;
  const int t     = threadIdx.x;
  const int lane  = t & 31;
  const int wv    = t >> 5;
  const int lmod  = lane & 15;
  const int lhalf = lane >> 4;

  f32x8 acc = {};
  for (int s0 = 0; s0 < S_DIM; s0 += 128) {
    // prefetch next s-tile: thread t covers row (s0+128+t), two 128B lines
    if (s0 + 128 < S_DIM) {
      const float* nxt = hidden + (size_t)(s0 + 128 + t) * BH + b * H_DIM + h0;
      __builtin_prefetch(nxt, 0, 1);
      __builtin_prefetch(nxt + 32, 0, 1);
    }
    __syncthreads();
#pragma unroll
    for (int i = 0; i < 4; ++i) {   // attn tile: 16 x 128
      const int idx = (i * 128 + t) * 4;
      const int n = idx >> 7, j = idx & 127;
      cp_b128(&lds_a[n * 132 + j],
              attn + (size_t)(b * NX_DIM + n) * S_DIM + s0 + j);
    }
#pragma unroll
    for (int i = 0; i < 16; ++i) {  // hidden tile: 128 x 64
      const int idx = (i * 128 + t) * 4;
      const int si = idx >> 6, hh = idx & 63;
      cp_b128(&lds_h[si * 72 + hh],
              hidden + (size_t)(s0 + si) * BH + b * H_DIM + h0 + hh);
    }
    cp_wait_barrier();
#pragma unroll 4
    for (int k0 = 0; k0 < 128; k0 += 4) {
      const int bk = k0 + 2 * lhalf;
      f32x2 a = *(const f32x2*)(&lds_a[lmod * 132 + bk]);
      f32x2 bb;
      bb.x = lds_h[bk * 72 + wv * 16 + lmod];
      bb.y = lds_h[(bk + 1) * 72 + wv * 16 + lmod];
      acc = wmma4(a, bb, acc);
    }
  }
#pragma unroll
  for (int r = 0; r < 8; ++r) {
    const int m = r + 8 * lhalf;
    aH[(size_t)(b * NX_DIM + m) * H_DIM + h0 + wv * 16 + lmod] = acc[r];
  }
}

// ---------------------------------------------------------------------------
// out[b,n,h] = aH[b,n,:] . V_w[h,:] + V_b[h]   for n in [0,16)
// grid (4, B), 128 threads.  (B-fragments are contiguous f32x2 from V_w rows.)
// ---------------------------------------------------------------------------
__global__ void k_qs(const float* __restrict__ aH, const float* __restrict__ Vw,
                     const float* __restrict__ Vb, float* __restrict__ out) {
  const int b     = blockIdx.y;
  const int h0    = blockIdx.x * 64;
  const int lane  = threadIdx.x & 31;
  const int wv    = threadIdx.x >> 5;
  const int lmod  = lane & 15;
  const int lhalf = lane >> 4;
  const int h     = h0 + wv * 16 + lmod;

  f32x8 acc = {};
  for (int k0 = 0; k0 < H_DIM; k0 += 4) {
    const int bk = k0 + 2 * lhalf;
    f32x2 a  = *(const f32x2*)(aH + (size_t)(b * NX_DIM + lmod) * H_DIM + bk);
    f32x2 bb = *(const f32x2*)(Vw + (size_t)h * H_DIM + bk);
    acc = wmma4(a, bb, acc);
  }
#pragma unroll
  for (int r = 0; r < 8; ++r) {
    const int m = r + 8 * lhalf;
    out[(size_t)b * (NX_DIM + 1) * H_DIM + m * H_DIM + h] = acc[r] + Vb[h];
  }
}

// ---------------------------------------------------------------------------
// Single collapsed LSTM step (h0=c0=0 => only i,g,o gates matter, W_hh unused)
// out[b,16,h] = sigmoid(zo) * tanh( sigmoid(zi) * tanh(zg) )
// grid (B), 256 threads (one per h).
// ---------------------------------------------------------------------------
__global__ void k_lstm(const float* __restrict__ Wih,
                       const float* __restrict__ bih,
                       const float* __restrict__ bhh,
                       float* __restrict__ out) {
  __shared__ float q[H_DIM];
  const int b = blockIdx.x;
  const int t = threadIdx.x;
  float* ob = out + (size_t)b * (NX_DIM + 1) * H_DIM;
  q[t] = ob[t];  // qs[b, 0, t] already written by k_qs
  __syncthreads();

  float zi = bih[t] + bhh[t];
  float zg = bih[2 * H_DIM + t] + bhh[2 * H_DIM + t];
  float zo = bih[3 * H_DIM + t] + bhh[3 * H_DIM + t];
  const float* wi = Wih + (size_t)t * H_DIM;
  const float* wg = Wih + (size_t)(2 * H_DIM + t) * H_DIM;
  const float* wo = Wih + (size_t)(3 * H_DIM + t) * H_DIM;
  for (int k = 0; k < H_DIM; ++k) {
    const float qv = q[k];
    zi = fmaf(qv, wi[k], zi);
    zg = fmaf(qv, wg[k], zg);
    zo = fmaf(qv, wo[k], zo);
  }
  const float ig = 1.f / (1.f + __expf(-zi));
  const float gg = tanhf(zg);
  const float og = 1.f / (1.f + __expf(-zo));
  const float c  = ig * gg;
  ob[NX_DIM * H_DIM + t] = og * tanhf(c);
}

// ---------------------------------------------------------------------------
extern "C" void kernel_launch(void* const* d_in, const int* in_sizes, int n_in,
                              void* d_out, int out_size, void* d_ws,
                              size_t ws_size, hipStream_t stream) {
  (void)in_sizes; (void)n_in; (void)out_size; (void)ws_size;
  const float* hidden = (const float*)d_in[0];  // [S,B,H]
  const float* x      = (const float*)d_in[1];  // [B,NX,H]
  const float* Kw     = (const float*)d_in[2];  // [H,H]
  const float* Kb     = (const float*)d_in[3];  // [H]
  const float* Vw     = (const float*)d_in[4];  // [H,H]
  const float* Vb     = (const float*)d_in[5];  // [H]
  const float* Wih    = (const float*)d_in[6];  // [4H,H]
  // d_in[7] = W_hh: provably unused (h0 == 0)
  const float* bih    = (const float*)d_in[8];  // [4H]
  const float* bhh    = (const float*)d_in[9];  // [4H]
  float* out = (float*)d_out;                   // [B, NX+1, H]

  // workspace layout (~5.2 MB)
  char* ws = (char*)d_ws;
  float* xK_ws  = (float*)(ws);                          // B*16*256
  float* xb_ws  = (float*)(ws + 524288);                 // B*16
  float* att_ws = (float*)(ws + 524288 + 4096);          // B*16*2048
  float* aH_ws  = (float*)(ws + 524288 + 4096 + 4194304);// B*16*256

  k_xk     <<<dim3(H_DIM / 64, B_DIM), 128, 0, stream>>>(x, Kw, xK_ws);
  k_xb     <<<dim3(B_DIM),             128, 0, stream>>>(x, Kb, xb_ws);
  k_scores <<<dim3(S_DIM / 64, B_DIM), 128, 0, stream>>>(hidden, xK_ws, xb_ws, att_ws);
  k_softmax<<<dim3(B_DIM * NX_DIM),    256, 0, stream>>>(att_ws);
  k_aH     <<<dim3(H_DIM / 64, B_DIM), 128, 0, stream>>>(hidden, att_ws, aH_ws);
  k_qs     <<<dim3(H_DIM / 64, B_DIM), 128, 0, stream>>>(aH_ws, Vw, Vb, out);
  k_lstm   <<<dim3(B_DIM),             256, 0, stream>>>(Wih, bih, bhh, out);
}